// GatingModule_88931592831412
// MI455X (gfx1250) — compile-verified
//
#include <hip/hip_runtime.h>
#include <math.h>
#include <stdint.h>

typedef float v2f __attribute__((ext_vector_type(2)));
typedef float v8f __attribute__((ext_vector_type(8)));

#define B_ROWS  32768
#define DDIM    4096
#define EXP     64
#define ROWS_WG 64
#define KC      64
#define NCH     (DDIM / KC)      // 64 K-chunks
#define LSTR    68               // padded LDS row stride (floats): 16B-aligned rows, conflict-free
#define ASZ     (ROWS_WG * LSTR) // one 64x64 staged array (padded)
#define BUFSZ   (3 * ASZ)        // x + w_gate + w_noise tile
#define SMEM_FLOATS (2 * BUFSZ)  // double buffer

// Async copy global -> LDS, no data VGPRs, tracked by ASYNCcnt.
// LDS byte address = low 32 bits of the generic shared pointer (flat LDS aperture).
__device__ __forceinline__ void async_b128(const void* gp, void* lp) {
    uint32_t lds = (uint32_t)(uintptr_t)lp;
    uint64_t ga  = (uint64_t)(uintptr_t)gp;
    asm volatile("global_load_async_to_lds_b128 %0, %1, off"
                 :: "v"(lds), "v"(ga) : "memory");
}
__device__ __forceinline__ void wait_async0() {
    asm volatile("s_wait_asynccnt 0" ::: "memory");
}

__global__ void gating_zero_load(float* __restrict__ loadp) {
    if (threadIdx.x < EXP) loadp[threadIdx.x] = 0.0f;
}

// Stage one 64x64 f32 chunk of x / w_gate / w_noise into a buffer (12 b128s/thread).
__device__ __forceinline__ void issue_chunk(const float* xb, const float* gb, const float* nb,
                                            float* L0, float* L1, float* L2, int t) {
    #pragma unroll
    for (int i = 0; i < 4; ++i) {
        int idx = t + i * 256;          // 0..1023 float4 slots
        int row = idx >> 4;             // 16 float4 per 64-float row
        int c4  = (idx & 15) << 2;
        int go  = row * DDIM + c4;
        int lo  = row * LSTR + c4;
        async_b128(xb + go, L0 + lo);
        async_b128(gb + go, L1 + lo);
        async_b128(nb + go, L2 + lo);
    }
}

__global__ __launch_bounds__(256) void gating_main(
    const float* __restrict__ x,  const float* __restrict__ wgate,
    const float* __restrict__ bg, const float* __restrict__ wnoise,
    const float* __restrict__ bn, const float* __restrict__ nz,
    float* __restrict__ out_gates, float* __restrict__ out_load)
{
    extern __shared__ float sm[];            // 2 * BUFSZ floats (dynamic: 104448 B)
    float* R0 = sm;                          // buf0 arr0: x tile -> clean logits -> prob
    float* R1 = sm + ASZ;                    // buf0 arr1: w_gate tile -> noisy logits -> gates
    float* R2 = sm + 2 * ASZ;                // buf0 arr2: w_noise tile -> noise tile -> stddev

    const int t    = threadIdx.x;
    const int lane = t & 31;                 // wave32
    const int w    = t >> 5;                 // 8 waves
    const int rt   = w & 3;                  // row tile 0..3 (16 rows each)
    const int mat  = w >> 2;                 // 0 = gate GEMM, 1 = noise GEMM
    const int r0   = blockIdx.x * ROWS_WG;

    const float* xB = x + (size_t)r0 * DDIM;

    v8f acc0 = {}, acc1 = {}, acc2 = {}, acc3 = {};

    const int m  = lane & 15;                // M (and N) index within tile
    const int kb = (lane >> 4) << 1;         // 0 or 2: K sub-offset per ISA f32 A/B layout

    // prefetch chunk 0 into buf0
    issue_chunk(xB, wgate, wnoise, sm, sm + ASZ, sm + 2 * ASZ, t);

    for (int c = 0; c < NCH; ++c) {
        float* cb = sm + (c & 1) * BUFSZ;

        wait_async0();                       // my async loads of chunk c are in LDS
        __syncthreads();                     // everyone's are

        if (c + 1 < NCH) {                   // overlap: next chunk into other buffer
            float* nb = sm + ((c + 1) & 1) * BUFSZ;
            const int k0 = (c + 1) * KC;
            issue_chunk(xB + k0, wgate + k0, wnoise + k0,
                        nb, nb + ASZ, nb + 2 * ASZ, t);
        } else {
            // last chunk: stage the 64x64 noise tile into buf0 arr2 (idle region)
            const float* nzB = nz + (size_t)r0 * EXP;
            #pragma unroll
            for (int i = 0; i < 4; ++i) {
                int idx = t + i * 256;
                int row = idx >> 4;
                int c4  = (idx & 15) << 2;
                async_b128(nzB + row * EXP + c4, R2 + row * LSTR + c4);
            }
        }

        const float* Ap = cb;
        const float* Wp = cb + (mat ? 2 : 1) * ASZ;
        #pragma unroll
        for (int ks = 0; ks < 16; ++ks) {
            const int kk = ks * 4 + kb;
            v2f a  = *(const v2f*)&Ap[(rt * 16 + m) * LSTR + kk];
            v2f b0 = *(const v2f*)&Wp[( 0 + m) * LSTR + kk];
            v2f b1 = *(const v2f*)&Wp[(16 + m) * LSTR + kk];
            v2f b2 = *(const v2f*)&Wp[(32 + m) * LSTR + kk];
            v2f b3 = *(const v2f*)&Wp[(48 + m) * LSTR + kk];
            acc0 = __builtin_amdgcn_wmma_f32_16x16x4_f32(false, a, false, b0, (short)0, acc0, false, false);
            acc1 = __builtin_amdgcn_wmma_f32_16x16x4_f32(false, a, false, b1, (short)0, acc1, false, false);
            acc2 = __builtin_amdgcn_wmma_f32_16x16x4_f32(false, a, false, b2, (short)0, acc2, false, false);
            acc3 = __builtin_amdgcn_wmma_f32_16x16x4_f32(false, a, false, b3, (short)0, acc3, false, false);
        }
    }

    wait_async0();                           // noise tile landed
    __syncthreads();                         // all compute done (buf0 arr0/1 free)

    // ---- write accumulators to LDS: R0 = clean logits, R1 = raw noise logits ----
    {
        float* Cd = mat ? R1 : R0;
        #pragma unroll
        for (int r = 0; r < 8; ++r) {
            int row = rt * 16 + ((lane >> 4) << 3) + r;  // C/D layout: M = 8*(lane/16)+r
            int col = lane & 15;
            Cd[row * LSTR + col +  0] = acc0[r];
            Cd[row * LSTR + col + 16] = acc1[r];
            Cd[row * LSTR + col + 32] = acc2[r];
            Cd[row * LSTR + col + 48] = acc3[r];
        }
    }
    __syncthreads();

    // ---- fused per-row epilogue: one thread per row ----
    if (t < ROWS_WG) {
        const int row = t;
        float vals[9];
        int   idxs[9];
        #pragma unroll
        for (int j = 0; j < 9; ++j) { vals[j] = -3.4e38f; idxs[j] = -1; }

        for (int e = 0; e < EXP; ++e) {
            float cl = R0[row * LSTR + e] + bg[e];
            float nr = R1[row * LSTR + e] + bn[e];
            float s  = (nr > 20.0f ? nr : log1pf(expf(nr))) + 0.01f;   // softplus + eps
            float ny = cl + R2[row * LSTR + e] * s;                    // noise from LDS
            R0[row * LSTR + e] = cl;
            R1[row * LSTR + e] = ny;
            R2[row * LSTR + e] = s;                                    // stddev overwrites noise
            if (ny > vals[8]) {               // top-9 insertion (descending)
                int j = 8;
                while (j > 0 && ny > vals[j - 1]) {
                    vals[j] = vals[j - 1]; idxs[j] = idxs[j - 1]; --j;
                }
                vals[j] = ny; idxs[j] = e;
            }
        }

        float mx = vals[0], se = 0.0f, g[8];
        #pragma unroll
        for (int j = 0; j < 8; ++j) { g[j] = expf(vals[j] - mx); se += g[j]; }
        const float inv  = 1.0f / se;
        const float thrI = vals[8];           // (K+1)-th value
        const float thrO = vals[7];           // K-th value

        for (int e = 0; e < EXP; ++e) {
            float cl  = R0[row * LSTR + e];
            float ny  = R1[row * LSTR + e];
            float s   = R2[row * LSTR + e];
            float thr = (ny > thrI) ? thrI : thrO;
            float z   = (cl - thr) / s;
            float p   = 0.5f * (1.0f + erff(z * 0.70710678118654752f));
            float gv  = 0.0f;
            #pragma unroll
            for (int j = 0; j < 8; ++j) gv = (idxs[j] == e) ? g[j] * inv : gv;
            R0[row * LSTR + e] = p;           // prob
            R1[row * LSTR + e] = gv;          // gate value
        }
    }
    __syncthreads();

    // ---- load = prob.sum(0): column sums, one global atomic per expert per WG ----
    if (t < EXP) {
        float s = 0.0f;
        for (int r = 0; r < ROWS_WG; ++r) s += R0[r * LSTR + t];
        atomicAdd(&out_load[t], s);
    }

    // ---- coalesced gates store ----
    #pragma unroll
    for (int i = 0; i < 16; ++i) {
        int idx = t + i * 256;                // 0..4095
        int row = idx >> 6;
        int e   = idx & 63;
        out_gates[(size_t)(r0 + row) * EXP + e] = R1[row * LSTR + e];
    }
}

extern "C" void kernel_launch(void* const* d_in, const int* in_sizes, int n_in,
                              void* d_out, int out_size, void* d_ws, size_t ws_size,
                              hipStream_t stream) {
    const float* x      = (const float*)d_in[0];
    const float* wgate  = (const float*)d_in[1];
    const float* bg     = (const float*)d_in[2];
    const float* wnoise = (const float*)d_in[3];
    const float* bn     = (const float*)d_in[4];
    const float* nz     = (const float*)d_in[5];

    float* out_gates = (float*)d_out;
    float* out_load  = out_gates + (size_t)B_ROWS * EXP;

    gating_zero_load<<<1, 64, 0, stream>>>(out_load);
    gating_main<<<B_ROWS / ROWS_WG, 256, SMEM_FLOATS * sizeof(float), stream>>>(
        x, wgate, bg, wnoise, bn, nz, out_gates, out_load);
}